// ChannelMaxPool_84293028151431
// MI455X (gfx1250) — compile-verified
//
#include <hip/hip_runtime.h>
#include <stdint.h>
#include <stddef.h>

typedef float f4 __attribute__((ext_vector_type(4)));

#define B_  32
#define C_  512
#define HW_ 16384   // 128*128 spatial
#define K_  128

// ---- CDNA5 async global->LDS path: device-only, inline asm (bypasses the
// ---- builtin's cuda_device-qualified pointer params that HIP can't spell) --
#if defined(__HIP_DEVICE_COMPILE__) && defined(__gfx1250__)
#define ASYNC_PATH 1
#define AS3 __attribute__((address_space(3)))
#ifndef __has_builtin
#define __has_builtin(x) 0
#endif
#if __has_builtin(__builtin_amdgcn_s_wait_asynccnt)
#define WAIT_ASYNC(n) do { __builtin_amdgcn_s_wait_asynccnt(n); asm volatile("" ::: "memory"); } while (0)
#else
#define WAIT_ASYNC(n) asm volatile("s_wait_asynccnt %0" :: "i"(n) : "memory")
#endif
#define ASYNC_LOAD_B128(ldsptr, gptr)                                          \
    asm volatile("global_load_async_to_lds_b128 %0, %1, off"                   \
                 :: "v"(ldsptr), "v"(gptr) : "memory")
#else
#define ASYNC_PATH 0
#endif

__device__ __forceinline__ float max_abs4(float m, f4 v) {
    return fmaxf(m, fmaxf(fmaxf(fabsf(v.x), fabsf(v.y)),
                          fmaxf(fabsf(v.z), fabsf(v.w))));
}

// ---------------------------------------------------------------------------
// Pass 1: scores[b,c] = max |x[b,c,:,:]|   (reads 1 GiB — the dominant cost)
// One 256-thread block (8 wave32) per (b,c); 16 f4 per thread, double-buffered
// through LDS with async global->LDS b128 loads (ASYNCcnt pipeline).
// ---------------------------------------------------------------------------
__global__ __launch_bounds__(256) void scores_kernel(const float* __restrict__ x,
                                                     float* __restrict__ scores) {
    const int c   = blockIdx.x;
    const int b   = blockIdx.y;
    const int tid = threadIdx.x;
    const f4* __restrict__ p = (const f4*)(x + ((size_t)b * C_ + c) * HW_);  // 4096 f4
    float m = 0.0f;

#if ASYNC_PATH
    __shared__ f4 stage[2][256];
    {   // prime the pipeline with chunk 0
        const f4* g0 = p + tid;
        AS3 f4*   l0 = (AS3 f4*)&stage[0][tid];
        ASYNC_LOAD_B128(l0, g0);
    }
    for (int chunk = 0; chunk < 15; ++chunk) {
        const f4* g = p + (chunk + 1) * 256 + tid;
        AS3 f4*   l = (AS3 f4*)&stage[(chunk + 1) & 1][tid];
        ASYNC_LOAD_B128(l, g);           // issue next chunk
        WAIT_ASYNC(1);                   // wait until previous chunk landed
        f4 v = stage[chunk & 1][tid];    // per-lane slot: no cross-thread share
        m = max_abs4(m, v);
    }
    WAIT_ASYNC(0);
    m = max_abs4(m, stage[1][tid]);      // chunk 15
#else
    #pragma unroll 4
    for (int chunk = 0; chunk < 16; ++chunk) {
        f4 v = p[chunk * 256 + tid];
        m = max_abs4(m, v);
    }
#endif

    // wave32 reduction, then combine the 8 waves through LDS
    #pragma unroll
    for (int off = 16; off > 0; off >>= 1)
        m = fmaxf(m, __shfl_xor(m, off, 32));
    __shared__ float wred[8];
    if ((tid & 31) == 0) wred[tid >> 5] = m;
    __syncthreads();
    if (tid == 0) {
        float r = wred[0];
        #pragma unroll
        for (int i = 1; i < 8; ++i) r = fmaxf(r, wred[i]);
        scores[(size_t)b * C_ + c] = r;
    }
}

// ---------------------------------------------------------------------------
// Pass 2: per batch, top-128 of 512 scores (desc, ties -> lower index) +
// renormalized softmax weights. One 512-thread block per batch; rank by
// exhaustive comparison (O(C^2)=256K cmp, trivial). Full-softmax denominator
// cancels under NORMALIZE_WEIGHTS, so weight_i = exp((s_i-m)/T)/sum_sel.
// ---------------------------------------------------------------------------
__global__ __launch_bounds__(512) void topk_kernel(const float* __restrict__ scores,
                                                   int* __restrict__ topk,
                                                   float* __restrict__ wt) {
    const int b = blockIdx.x;
    const int t = threadIdx.x;
    __shared__ float s[C_];
    __shared__ float sel[K_];
    __shared__ float red[16];

    s[t] = scores[(size_t)b * C_ + t];
    __syncthreads();

    const float v = s[t];
    int rank = 0;
    #pragma unroll 8
    for (int j = 0; j < C_; ++j) {
        const float u = s[j];
        rank += (u > v) || (u == v && j < t);   // jax top_k tie-break
    }
    if (rank < K_) {
        topk[b * K_ + rank] = t;
        sel[rank] = v;
    }
    __syncthreads();

    const float m     = sel[0];                 // rank 0 == global max
    const float inv_t = 1.0f / (1.0f + 1e-12f); // TEMPERATURE + 1e-12
    float w = 0.0f;
    if (t < K_) w = expf((sel[t] - m) * inv_t);

    float sum = w;                              // block-sum (lanes >= K add 0)
    #pragma unroll
    for (int off = 16; off > 0; off >>= 1)
        sum += __shfl_xor(sum, off, 32);
    if ((t & 31) == 0) red[t >> 5] = sum;
    __syncthreads();
    if (t == 0) {
        float acc = 0.0f;
        #pragma unroll
        for (int i = 0; i < 16; ++i) acc += red[i];
        red[0] = fmaxf(acc, 1e-12f);
    }
    __syncthreads();
    if (t < K_) wt[b * K_ + t] = w / red[0];
}

// ---------------------------------------------------------------------------
// Pass 3: out[b,k,:,:] = x[b, idx[b,k], :, :] * wt[b,k]
// b128 streaming loads + non-temporal b128 stores (output is write-once; keep
// the 192MB L2 for input lines pass 1 warmed). Prefetch ahead.
// ---------------------------------------------------------------------------
__global__ __launch_bounds__(256) void gather_kernel(const float* __restrict__ x,
                                                     const int* __restrict__ topk,
                                                     const float* __restrict__ wt,
                                                     float* __restrict__ out) {
    const int k = blockIdx.x;
    const int b = blockIdx.y;
    const int c = topk[b * K_ + k];
    const float w = wt[b * K_ + k];
    const f4* __restrict__ src = (const f4*)(x   + ((size_t)b * C_ + c) * HW_);
    f4* __restrict__       dst = (f4*)      (out + ((size_t)b * K_ + k) * HW_);
    #pragma unroll 4
    for (int i = threadIdx.x; i < HW_ / 4; i += 256) {
        if (i + 1024 < HW_ / 4) __builtin_prefetch(&src[i + 1024], 0, 0);
        f4 v = src[i];
        v *= w;
        __builtin_nontemporal_store(v, &dst[i]);
    }
}

// ---------------------------------------------------------------------------
extern "C" void kernel_launch(void* const* d_in, const int* in_sizes, int n_in,
                              void* d_out, int out_size, void* d_ws, size_t ws_size,
                              hipStream_t stream) {
    (void)in_sizes; (void)n_in; (void)out_size; (void)ws_size;
    const float* x = (const float*)d_in[0];
    float* out = (float*)d_out;

    // scratch: scores (64 KiB) | topk idx (16 KiB) | weights (16 KiB)
    float* scores = (float*)d_ws;
    int*   topk   = (int*)  ((char*)d_ws + (size_t)B_ * C_ * sizeof(float));
    float* wt     = (float*)((char*)d_ws + (size_t)B_ * C_ * sizeof(float)
                                         + (size_t)B_ * K_ * sizeof(int));

    scores_kernel<<<dim3(C_, B_), 256, 0, stream>>>(x, scores);
    topk_kernel  <<<B_, C_, 0, stream>>>(scores, topk, wt);
    gather_kernel<<<dim3(K_, B_), 256, 0, stream>>>(x, topk, wt, out);
}